// KernelConv_60473139528286
// MI455X (gfx1250) — compile-verified
//
#include <hip/hip_runtime.h>
#include <math.h>

// ---------------------------------------------------------------------------
// Equivariant tensor-product convolution (e3nn-style) for MI455X / gfx1250.
// Restructured: sum over neighbors commutes with the 2nd MLP layer:
//   B[z,y,pw] = (sum_a Yext[z,a,y] * h(z,a,:)) @ w2 + (sum_a Yext[z,a,y]) * b2
// y = 0..8 -> Y components (nz pairs), y = 9 -> Y0 of self (!nz) pairs.
// All GEMMs use V_WMMA_F32_16X16X4_F32 (full f32 precision; problem is
// memory-bound at ~10 GFLOPs vs 23.3 TB/s HBM, so f32 WMMA is the ceiling-
// appropriate choice).
// ---------------------------------------------------------------------------

typedef float v2f __attribute__((ext_vector_type(2)));
typedef float v8f __attribute__((ext_vector_type(8)));

#define WMMA_F32(A, B, C) \
    __builtin_amdgcn_wmma_f32_16x16x4_f32(false, (A), false, (B), (short)0, (C), false, false)

// ---- constants ------------------------------------------------------------
#define ZTOT   1024     // b*n
#define NNB    64       // neighbors per point
#define NRBF   64
#define HID    256
#define NPATH  1536

#define Y0C      0.28209479177387814f
#define C1SH     0.4886025119029199f
#define CA2      1.0925484305920792f
#define CB2      0.31539156525252005f
#define CC2      0.5462742152960396f

// norm coefficients: nc[i][j][0] = sqrt(2*l_in+1)*sqrt(4pi)/sqrt(num_i),
//                    nc[i][j][1] = ... /sqrt(mul_in)
#define NC00_0   0.6266570686577501f   // sqrt(4pi)/sqrt(32)
#define NC00_1   0.8862269254527580f   // sqrt(4pi)/4
#define NC01_0   1.0854019463332790f   // sqrt(3)*sqrt(4pi)/sqrt(32) = sqrt(3pi/8)
#define NC10_0   0.4431134627263790f   // sqrt(4pi)/8
#define NC11_0   0.7674950309566161f   // sqrt(3)*sqrt(4pi)/8
#define NC11_1   1.5349900619132322f   // sqrt(3)*sqrt(4pi)/4

#define INV_SQRT3 0.5773502691896258f
#define INV_SQRT6 0.4082482904638630f
#define K10       0.3162277660168379f  // 1/sqrt(10)
#define K30       0.1825741858350554f  // 1/sqrt(30)

// real CG for (l_out=1, l_in=1, lf=2), index [a][b][c], c over [xy,yz,m0,xz,x2-y2]
__constant__ float C2T[3][3][5] = {
    { {0.f, 0.f, -K30, 0.f, -K10}, {0.f, K10, 0.f, 0.f, 0.f}, {K10, 0.f, 0.f, 0.f, 0.f} },
    { {0.f, K10, 0.f, 0.f, 0.f},   {0.f, 0.f, 2.f*K30, 0.f, 0.f}, {0.f, 0.f, 0.f, K10, 0.f} },
    { {K10, 0.f, 0.f, 0.f, 0.f},   {0.f, 0.f, 0.f, K10, 0.f}, {0.f, 0.f, -K30, 0.f, K10} },
};

// ---------------------------------------------------------------------------
// Kernel 1: per z, H = relu(RBF @ w1 + b1)  (64x256), G = Yext^T @ H (16x256)
// ---------------------------------------------------------------------------
__global__ __launch_bounds__(256) void tp_pair_kernel(
    const float* __restrict__ geom,   // (16,64,3)
    const float* __restrict__ w1,     // (64,256)
    const float* __restrict__ b1,     // (256)
    float* __restrict__ G,            // (Z,16,256)
    float* __restrict__ S)            // (Z,16)
{
    __shared__ float RB[64][68];      // rbf[a][r], padded stride
    __shared__ float Yx[16][68];      // Yext[y][a]
    __shared__ float Hs[64][128];     // H half-tile (a x 128 hidden cols)
    __shared__ float gx[64], gy[64], gz[64], rad[64];

    const int z   = blockIdx.x;
    const int bi  = z >> 6;
    const int pi  = z & 63;
    const int tid = threadIdx.x;

    if (tid < 64) {
        const float* gp = geom + (bi * 64 + tid) * 3;
        gx[tid] = gp[0]; gy[tid] = gp[1]; gz[tid] = gp[2];
    }
    __syncthreads();

    if (tid < 64) {
        const int a = tid;
        float dx = gx[pi] - gx[a], dy = gy[pi] - gy[a], dz = gz[pi] - gz[a];
        float d2 = dx*dx + dy*dy + dz*dz;
        bool  nz = d2 > 0.f;
        float r  = nz ? sqrtf(d2) : 0.f;
        float iv = nz ? 1.f / r : 0.f;
        float ux = dx * iv, uy = dy * iv, uz = dz * iv;
        float m  = nz ? 1.f : 0.f;
        rad[a] = r;
        Yx[0][a] = m * Y0C;                    // Y0 for nz pairs
        Yx[1][a] = C1SH * uy;                  // l=1 (zero when !nz: u=0)
        Yx[2][a] = C1SH * uz;
        Yx[3][a] = C1SH * ux;
        Yx[4][a] = CA2 * ux * uy;              // l=2
        Yx[5][a] = CA2 * uy * uz;
        Yx[6][a] = m * CB2 * (3.f * uz * uz - 1.f);
        Yx[7][a] = CA2 * ux * uz;
        Yx[8][a] = CC2 * (ux * ux - uy * uy);
        Yx[9][a] = (1.f - m) * Y0C;            // self row
        for (int yy = 10; yy < 16; ++yy) Yx[yy][a] = 0.f;
    }
    __syncthreads();

    // RBF matrix: RB[a][r] = exp(-10*(rad[a] - 3*r/63)^2)
    for (int idx = tid; idx < 64 * 64; idx += 256) {
        int a = idx >> 6, rr = idx & 63;
        float d = rad[a] - (3.0f / 63.0f) * (float)rr;
        RB[a][rr] = expf(-10.0f * d * d);
    }
    __syncthreads();

    const int wave = tid >> 5;
    const int lane = tid & 31;
    const int nl   = lane & 15;       // N / M lane index
    const int kh   = lane >> 4;       // K half select

    // S[z][yy] = sum_a Yext[yy][a]
    if (tid < 16) {
        float s = 0.f;
        for (int a = 0; a < 64; ++a) s += Yx[tid][a];
        S[z * 16 + tid] = s;
    }

    for (int h = 0; h < 2; ++h) {
        // GEMM1: H(64 x 128cols) = RB(64x64) @ w1(64, cols), 4 Mtiles x 8 Ntiles
        for (int t = wave; t < 32; t += 8) {
            int mt  = t >> 3;
            int col = (h * 8 + (t & 7)) * 16 + nl;   // global hidden column
            v8f acc = {};
            for (int ks = 0; ks < 16; ++ks) {
                int k0 = ks * 4 + 2 * kh;
                v2f af = { RB[mt * 16 + nl][k0], RB[mt * 16 + nl][k0 + 1] };
                v2f bf = { w1[k0 * HID + col], w1[(k0 + 1) * HID + col] };
                acc = WMMA_F32(af, bf, acc);
            }
            float bias = b1[col];
            int lcol = col - h * 128;
#pragma unroll
            for (int r = 0; r < 8; ++r) {
                float hv = acc[r] + bias;
                Hs[mt * 16 + r + 8 * kh][lcol] = hv > 0.f ? hv : 0.f;
            }
        }
        __syncthreads();

        // GEMM2: G(16 x 128cols) = Yx(16x64) @ Hs(64 x cols), one Ntile/wave
        {
            int col  = (h * 8 + wave) * 16 + nl;
            int lcol = col - h * 128;
            v8f acc = {};
            for (int ks = 0; ks < 16; ++ks) {
                int k0 = ks * 4 + 2 * kh;
                v2f af = { Yx[nl][k0], Yx[nl][k0 + 1] };
                v2f bf = { Hs[k0][lcol], Hs[k0 + 1][lcol] };
                acc = WMMA_F32(af, bf, acc);
            }
#pragma unroll
            for (int r = 0; r < 8; ++r)
                G[(z * 16 + r + 8 * kh) * HID + col] = acc[r];
        }
        __syncthreads();
    }
}

// ---------------------------------------------------------------------------
// Kernel 2: per z, B = G @ w2 + S x b2 (10 x 1536), then CG contraction to out
// Path layout: P0 pw 0..255 (l0xl0->l0), P1 256..511 (l0 out,l1 in,lf1),
//              P2 512..767 (l1 out,l0 in,lf1), P3 768..1535 (l1xl1,lf=0,1,2)
// ---------------------------------------------------------------------------
__global__ __launch_bounds__(256) void tp_out_kernel(
    const float* __restrict__ G,      // (Z,16,256)
    const float* __restrict__ S,      // (Z,16)
    const float* __restrict__ feats,  // (Z,64)
    const float* __restrict__ w2,     // (256,1536)
    const float* __restrict__ b2,     // (1536)
    float* __restrict__ out)          // (Z,64)
{
    __shared__ float Gl[16][260];
    __shared__ float Bl[10][768];
    __shared__ float fl[64];
    __shared__ float Sl[16];

    const int z   = blockIdx.x;
    const int tid = threadIdx.x;

    for (int idx = tid; idx < 16 * 256; idx += 256)
        Gl[idx >> 8][idx & 255] = G[z * (16 * 256) + idx];
    if (tid < 16) Sl[tid] = S[z * 16 + tid];
    if (tid < 64) fl[tid] = feats[z * 64 + tid];
    __syncthreads();

    const int wave = tid >> 5;
    const int lane = tid & 31;
    const int nl   = lane & 15;
    const int kh   = lane >> 4;

    float accOut = 0.f;

    for (int ph = 0; ph < 2; ++ph) {
        // GEMM: 48 N-tiles per phase (cols ph*768 .. ph*768+767), 6 per wave
        for (int q = 0; q < 6; ++q) {
            int nt  = ph * 48 + wave * 6 + q;
            int col = nt * 16 + nl;               // global path-weight column
            v8f acc = {};
            for (int ks = 0; ks < 64; ++ks) {
                int k0 = ks * 4 + 2 * kh;
                v2f af = { Gl[nl][k0], Gl[nl][k0 + 1] };
                v2f bf = { w2[k0 * NPATH + col], w2[(k0 + 1) * NPATH + col] };
                acc = WMMA_F32(af, bf, acc);
            }
            float bb   = b2[col];
            int   lcol = col - ph * 768;
#pragma unroll
            for (int r = 0; r < 8; ++r) {
                int yy = r + 8 * kh;
                if (yy < 10) Bl[yy][lcol] = acc[r] + Sl[yy] * bb;
            }
        }
        __syncthreads();

        if (ph == 0) {
            if (tid < 16) {                       // outputs 0..15 (l_out=0), u=tid
                const int u = tid;
                for (int v = 0; v < 16; ++v) {
                    int pw = u * 16 + v;
                    // P0: lf=0, C=1, nz + self rows
                    accOut += (NC00_0 * Bl[0][pw] + NC00_1 * Bl[9][pw]) * fl[v];
                    // P1: lf=1, C = delta/sqrt(3)
                    int p1 = 256 + pw;
                    float s = 0.f;
                    for (int m = 0; m < 3; ++m) s += Bl[1 + m][p1] * fl[16 + v * 3 + m];
                    accOut += INV_SQRT3 * NC01_0 * s;
                }
            } else if (tid < 64) {                // outputs 16..63 (l_out=1)
                const int idx = tid - 16;
                const int u = idx / 3, a = idx % 3;
                // P2: lf=1, C[a,0,c] = delta_ac/sqrt(3)
                for (int v = 0; v < 16; ++v)
                    accOut += INV_SQRT3 * NC10_0 * Bl[1 + a][512 + u * 16 + v] * fl[v];
            }
        } else {
            if (tid >= 16 && tid < 64) {          // P3: l1 x l1, lfs = (0,1,2)
                const int idx = tid - 16;
                const int u = idx / 3, a = idx % 3;
                const int bI = (a + 1) % 3, cI = (a + 2) % 3;
                for (int v = 0; v < 16; ++v) {
                    int p = (u * 16 + v) * 3;     // local (768 already subtracted)
                    // lf=0: C[a,b] = -delta_ab/sqrt(3); nz + self rows
                    accOut += (-INV_SQRT3) * (NC11_0 * Bl[0][p] + NC11_1 * Bl[9][p])
                              * fl[16 + v * 3 + a];
                    // lf=1: C[a,b,c] = -eps_abc/sqrt(6)
                    accOut += (-INV_SQRT6) * NC11_0 *
                              (Bl[1 + cI][p + 1] * fl[16 + v * 3 + bI] -
                               Bl[1 + bI][p + 1] * fl[16 + v * 3 + cI]);
                    // lf=2: symmetric traceless table
                    float t2 = 0.f;
                    for (int b = 0; b < 3; ++b)
                        for (int c = 0; c < 5; ++c)
                            t2 += C2T[a][b][c] * Bl[4 + c][p + 2] * fl[16 + v * 3 + b];
                    accOut += NC11_0 * t2;
                }
            }
        }
        __syncthreads();
    }

    if (tid < 64) out[z * 64 + tid] = accOut;
}

// ---------------------------------------------------------------------------
extern "C" void kernel_launch(void* const* d_in, const int* in_sizes, int n_in,
                              void* d_out, int out_size, void* d_ws, size_t ws_size,
                              hipStream_t stream) {
    (void)in_sizes; (void)n_in; (void)out_size; (void)ws_size;
    const float* feats = (const float*)d_in[0];
    const float* geom  = (const float*)d_in[1];
    const float* w1    = (const float*)d_in[2];
    const float* b1    = (const float*)d_in[3];
    const float* w2    = (const float*)d_in[4];
    const float* b2    = (const float*)d_in[5];
    float* out = (float*)d_out;

    float* G = (float*)d_ws;                       // Z*16*256 floats = 16 MB
    float* S = G + (size_t)ZTOT * 16 * HID;        // Z*16 floats

    tp_pair_kernel<<<ZTOT, 256, 0, stream>>>(geom, w1, b1, G, S);
    tp_out_kernel<<<ZTOT, 256, 0, stream>>>(G, S, feats, w2, b2, out);
}